// SCSA_42752104464793
// MI455X (gfx1250) — compile-verified
//
#include <hip/hip_runtime.h>
#include <math.h>

// ---------------------------------------------------------------------------
// SCSA (spatial + channel self-attention) for MI455X / gfx1250, wave32.
//
// Pipeline (all on `stream`):
//   K1  row/col means of x              (streams x: ~103MB read)
//   K2  dwconv(k=3,5,7,9)+GN+sigmoid    (tiny) -> Ah, Aw   [run twice]
//   K3  gated 8x8 pool                  (x re-read, L2-resident) -> xp (B,C,49)
//   K4  GN(groups=1) + channel attention via V_WMMA_F32_16X16X4_F32,
//       online softmax in registers; second GEMM collapsed to
//       Mc[c] = sigmoid( (sum_d p[c,d]*vs[d]) / 49 ),  vs[d]=sum_n v[d,n]
//   K5  out = x * Ah * Aw * Mc          (x re-read + 103MB write)
// Bandwidth-bound: ~206MB HBM floor -> ~9us at 23.3TB/s; attention is
// ~0.4 GFLOP so f32 WMMA (no precision loss) is the right choice.
// ---------------------------------------------------------------------------

typedef __attribute__((ext_vector_type(2))) float v2f;
typedef __attribute__((ext_vector_type(8))) float v8f;

#define BB   32
#define CC   256
#define HH   56
#define WW   56
#define CG   64
#define PLANE (HH * WW)        // 3136
#define NP   49                // 7*7
#define NPAD 52                // 13 * 4 (K padded for 16x16x4 steps)
#define EPS  1e-5f

// ---------------- K1: x_h = mean over W, x_w = mean over H -----------------
__global__ __launch_bounds__(256) void k_pool_hw(
    const float* __restrict__ x, float* __restrict__ xh, float* __restrict__ xw) {
  __shared__ float rowsum[HH];
  __shared__ float colsum[WW];
  const int bc  = blockIdx.x;
  const int tid = threadIdx.x;
  if (tid < HH) { rowsum[tid] = 0.f; colsum[tid] = 0.f; }
  __syncthreads();
  const float4* p = (const float4*)(x + (size_t)bc * PLANE);
  for (int idx = tid; idx < PLANE / 4; idx += 256) {  // 784 float4
    const int h = idx / 14, w4 = idx % 14;
    float4 v = p[idx];
    atomicAdd(&rowsum[h], v.x + v.y + v.z + v.w);
    atomicAdd(&colsum[w4 * 4 + 0], v.x);
    atomicAdd(&colsum[w4 * 4 + 1], v.y);
    atomicAdd(&colsum[w4 * 4 + 2], v.z);
    atomicAdd(&colsum[w4 * 4 + 3], v.w);
  }
  __syncthreads();
  if (tid < HH) {
    xh[(size_t)bc * HH + tid] = rowsum[tid] * (1.0f / (float)WW);
    xw[(size_t)bc * WW + tid] = colsum[tid] * (1.0f / (float)HH);
  }
}

// ------- K2: depthwise conv1d (same pad) + GroupNorm(4) + sigmoid ----------
// grid = B*4 (one block per (batch, group)); group g uses kernel k = 3+2g.
__global__ __launch_bounds__(256) void k_conv_gn_sig(
    const float* __restrict__ xin, const float* __restrict__ w0,
    const float* __restrict__ w1, const float* __restrict__ w2,
    const float* __restrict__ w3, const float* __restrict__ gw,
    const float* __restrict__ gb, float* __restrict__ out) {
  __shared__ float ybuf[CG * HH];   // 3584 conv outputs
  __shared__ float red1[256];
  __shared__ float red2[256];
  const int blk = blockIdx.x;
  const int b = blk >> 2, g = blk & 3;
  const int tid = threadIdx.x;
  const float* wg = (g == 0) ? w0 : (g == 1) ? w1 : (g == 2) ? w2 : w3;
  const int ks = 3 + 2 * g;
  const int kh = ks >> 1;
  const float* xg = xin + ((size_t)b * CC + (size_t)g * CG) * HH;
  float s = 0.f, s2 = 0.f;
  for (int i = tid; i < CG * HH; i += 256) {
    const int cl = i / HH, h = i % HH;
    const float* xr = xg + cl * HH;
    const float* wr = wg + cl * ks;
    float acc = 0.f;
    for (int t = 0; t < ks; ++t) {
      const int hh = h + t - kh;
      if (hh >= 0 && hh < HH) acc += xr[hh] * wr[t];
    }
    ybuf[i] = acc;
    s += acc; s2 += acc * acc;
  }
  red1[tid] = s; red2[tid] = s2;
  __syncthreads();
  for (int off = 128; off > 0; off >>= 1) {
    if (tid < off) { red1[tid] += red1[tid + off]; red2[tid] += red2[tid + off]; }
    __syncthreads();
  }
  const float inv_n = 1.0f / (float)(CG * HH);
  const float mu  = red1[0] * inv_n;
  const float var = red2[0] * inv_n - mu * mu;
  const float rs  = rsqrtf(var + EPS);
  float* og = out + ((size_t)b * CC + (size_t)g * CG) * HH;
  for (int i = tid; i < CG * HH; i += 256) {
    const int c = g * CG + i / HH;
    const float y = (ybuf[i] - mu) * rs * gw[c] + gb[c];
    og[i] = 1.0f / (1.0f + __expf(-y));
  }
}

// ---------------- K3: gated 8x8 average pool -> xp (B,C,49) ----------------
__global__ __launch_bounds__(256) void k_pool_pp(
    const float* __restrict__ x, const float* __restrict__ Ah,
    const float* __restrict__ Aw, float* __restrict__ xp) {
  __shared__ float bins[NP];
  __shared__ float ah_s[HH];
  __shared__ float aw_s[WW];
  const int bc = blockIdx.x;
  const int tid = threadIdx.x;
  if (tid < NP) bins[tid] = 0.f;
  if (tid < HH) { ah_s[tid] = Ah[(size_t)bc * HH + tid];
                  aw_s[tid] = Aw[(size_t)bc * WW + tid]; }
  __syncthreads();
  const float4* p = (const float4*)(x + (size_t)bc * PLANE);
  for (int idx = tid; idx < PLANE / 4; idx += 256) {
    const int h = idx / 14, w4 = idx % 14;
    float4 v = p[idx];
    const float contrib = (v.x * aw_s[w4 * 4 + 0] + v.y * aw_s[w4 * 4 + 1] +
                           v.z * aw_s[w4 * 4 + 2] + v.w * aw_s[w4 * 4 + 3]) * ah_s[h];
    atomicAdd(&bins[(h >> 3) * 7 + (w4 >> 1)], contrib);
  }
  __syncthreads();
  if (tid < NP) xp[(size_t)bc * NP + tid] = bins[tid] * (1.0f / 64.0f);
}

// --- K4: GN(groups=1) + channel attention with V_WMMA_F32_16X16X4_F32 ------
// One block per batch, 256 threads = 8 waves. Wave w owns row-blocks {w, w+8}.
__global__ __launch_bounds__(256) void k_attn(
    const float* __restrict__ xp, const float* __restrict__ gn1w,
    const float* __restrict__ gn1b, const float* __restrict__ wq,
    const float* __restrict__ wk, const float* __restrict__ wv,
    float* __restrict__ Mc) {
  __shared__ float xn[CC * NPAD];   // normalized xp, K-padded to 52 (53KB)
  __shared__ float vs[CC];          // vs[d] = wv[d] * sum_n xn[d,n]
  __shared__ float redbuf[16];
  __shared__ float stat[2];
  const int b   = blockIdx.x;
  const int tid = threadIdx.x;
  const float* src = xp + (size_t)b * CC * NP;

  // -- mean / var over all C*49 values (groups == 1) --
  float s = 0.f, s2 = 0.f;
  for (int i = tid; i < CC * NP; i += 256) { const float v = src[i]; s += v; s2 += v * v; }
#pragma unroll
  for (int off = 16; off >= 1; off >>= 1) {
    s  += __shfl_xor(s,  off, 32);
    s2 += __shfl_xor(s2, off, 32);
  }
  const int wid = tid >> 5, lane = tid & 31;
  if (lane == 0) { redbuf[wid] = s; redbuf[8 + wid] = s2; }
  __syncthreads();
  if (tid == 0) {
    float ts = 0.f, ts2 = 0.f;
    for (int i = 0; i < 8; ++i) { ts += redbuf[i]; ts2 += redbuf[8 + i]; }
    const float inv_n = 1.0f / (float)(CC * NP);
    const float mu  = ts * inv_n;
    const float var = ts2 * inv_n - mu * mu;
    stat[0] = mu; stat[1] = rsqrtf(var + EPS);
  }
  __syncthreads();
  const float mu = stat[0], rs = stat[1];

  // -- normalize into LDS (affine), zero the K padding --
  for (int i = tid; i < CC * NP; i += 256) {
    const int c = i / NP, n = i % NP;
    xn[c * NPAD + n] = (src[i] - mu) * rs * gn1w[c] + gn1b[c];
  }
  for (int i = tid; i < CC * (NPAD - NP); i += 256) {
    const int c = i / (NPAD - NP), j = i % (NPAD - NP);
    xn[c * NPAD + NP + j] = 0.f;
  }
  __syncthreads();
  {
    const int d = tid;                 // exactly 256 threads
    float a = 0.f;
    for (int n = 0; n < NP; ++n) a += xn[d * NPAD + n];
    vs[d] = a * wv[d];
  }
  __syncthreads();

  // -- s = q k^T / 7 via WMMA f32 16x16x4; online softmax; Mc = sig(S2/S1/49)
  const int hi = lane >> 4, lo = lane & 15;
  const float invSqrtN = 1.0f / 7.0f;
  for (int rbi = 0; rbi < 2; ++rbi) {
    const int c0 = (wid + rbi * 8) * 16;
    const float wqv = wq[c0 + lo];                 // A-row scale (M = lo)
    float m[8], S1[8], S2[8];
#pragma unroll
    for (int r = 0; r < 8; ++r) { m[r] = -INFINITY; S1[r] = 0.f; S2[r] = 0.f; }
    for (int cb = 0; cb < 16; ++cb) {
      const int d0 = cb * 16;
      const float wkv = wk[d0 + lo];               // B-col scale (N = lo)
      const float vsv = vs[d0 + lo];
      v8f acc = {};
      for (int kk = 0; kk < 13; ++kk) {
        const int kbase = kk * 4 + hi * 2;         // A/B per-lane K pair
        v2f A, Bv;
        A.x  = xn[(c0 + lo) * NPAD + kbase + 0] * wqv;
        A.y  = xn[(c0 + lo) * NPAD + kbase + 1] * wqv;
        Bv.x = xn[(d0 + lo) * NPAD + kbase + 0] * wkv;   // B[k][n] = k^T
        Bv.y = xn[(d0 + lo) * NPAD + kbase + 1] * wkv;
        acc = __builtin_amdgcn_wmma_f32_16x16x4_f32(
            false, A, false, Bv, (short)0, acc, false, false);
      }
      // D layout: reg r holds row (c0 + r + 8*hi), col (d0 + lo).
#pragma unroll
      for (int r = 0; r < 8; ++r) {
        const float xv = acc[r] * invSqrtN;
        float tm = xv;
#pragma unroll
        for (int mk = 8; mk >= 1; mk >>= 1) tm = fmaxf(tm, __shfl_xor(tm, mk, 16));
        const float mnew = fmaxf(m[r], tm);
        const float e = __expf(xv - mnew);
        float se = e, sev = e * vsv;
#pragma unroll
        for (int mk = 8; mk >= 1; mk >>= 1) {
          se  += __shfl_xor(se,  mk, 16);
          sev += __shfl_xor(sev, mk, 16);
        }
        const float scale = __expf(m[r] - mnew);
        S1[r] = S1[r] * scale + se;
        S2[r] = S2[r] * scale + sev;
        m[r] = mnew;
      }
    }
    if (lo == 0) {
#pragma unroll
      for (int r = 0; r < 8; ++r) {
        const int c = c0 + r + 8 * hi;
        const float z = S2[r] / (S1[r] * (float)NP);
        Mc[(size_t)b * CC + c] = 1.0f / (1.0f + __expf(-z));
      }
    }
  }
}

// ---------------- K5: out = x * Ah * Aw * Mc -------------------------------
__global__ __launch_bounds__(256) void k_finalize(
    const float* __restrict__ x, const float* __restrict__ Ah,
    const float* __restrict__ Aw, const float* __restrict__ Mc,
    float* __restrict__ out) {
  const size_t i4 = (size_t)blockIdx.x * 256 + threadIdx.x;  // float4 index
  const int w4 = (int)(i4 % 14);
  const size_t r = i4 / 14;
  const int h = (int)(r % HH);
  const size_t bc = r / HH;
  float4 xv  = ((const float4*)x)[i4];
  float4 awv = ((const float4*)(Aw + bc * WW))[w4];
  const float sc = Ah[bc * HH + h] * Mc[bc];
  float4 o;
  o.x = xv.x * awv.x * sc;
  o.y = xv.y * awv.y * sc;
  o.z = xv.z * awv.z * sc;
  o.w = xv.w * awv.w * sc;
  ((float4*)out)[i4] = o;
}

// ---------------------------------------------------------------------------
extern "C" void kernel_launch(void* const* d_in, const int* in_sizes, int n_in,
                              void* d_out, int out_size, void* d_ws, size_t ws_size,
                              hipStream_t stream) {
  // setup_inputs() dict order:
  const float* x     = (const float*)d_in[0];
  const float* dwh0  = (const float*)d_in[1];
  const float* dww0  = (const float*)d_in[2];
  const float* dwh1  = (const float*)d_in[3];
  const float* dww1  = (const float*)d_in[4];
  const float* dwh2  = (const float*)d_in[5];
  const float* dww2  = (const float*)d_in[6];
  const float* dwh3  = (const float*)d_in[7];
  const float* dww3  = (const float*)d_in[8];
  const float* gnh_w = (const float*)d_in[9];
  const float* gnh_b = (const float*)d_in[10];
  const float* gnw_w = (const float*)d_in[11];
  const float* gnw_b = (const float*)d_in[12];
  const float* gn1_w = (const float*)d_in[13];
  const float* gn1_b = (const float*)d_in[14];
  const float* wq    = (const float*)d_in[15];
  const float* wk    = (const float*)d_in[16];
  const float* wv    = (const float*)d_in[17];

  float* ws = (float*)d_ws;                 // ~8.6 MB of scratch used
  const size_t NBC  = (size_t)BB * CC;      // 8192
  float* xh = ws;                           // (B,C,H)
  float* xw = xh + NBC * HH;                // (B,C,W)
  float* Ah = xw + NBC * WW;                // (B,C,H)
  float* Aw = Ah + NBC * HH;                // (B,C,W)
  float* xp = Aw + NBC * WW;                // (B,C,49)
  float* Mc = xp + NBC * NP;                // (B,C)

  k_pool_hw<<<(int)NBC, 256, 0, stream>>>(x, xh, xw);
  k_conv_gn_sig<<<BB * 4, 256, 0, stream>>>(xh, dwh0, dwh1, dwh2, dwh3,
                                            gnh_w, gnh_b, Ah);
  k_conv_gn_sig<<<BB * 4, 256, 0, stream>>>(xw, dww0, dww1, dww2, dww3,
                                            gnw_w, gnw_b, Aw);
  k_pool_pp<<<(int)NBC, 256, 0, stream>>>(x, Ah, Aw, xp);
  k_attn<<<BB, 256, 0, stream>>>(xp, gn1_w, gn1_b, wq, wk, wv, Mc);
  const int nf4 = (BB * CC * HH * WW) / 4;  // 6,422,528 float4
  k_finalize<<<nf4 / 256, 256, 0, stream>>>(x, Ah, Aw, Mc, (float*)d_out);
}